// RACNN_33870112096712
// MI455X (gfx1250) — compile-verified
//
#include <hip/hip_runtime.h>

#define SS 224
#define OUTN 224

typedef float v2f __attribute__((ext_vector_type(2)));
typedef float v8f __attribute__((ext_vector_type(8)));

__device__ __forceinline__ float sig10(float z) {
    // sigmoid(10*z)
    return 1.0f / (1.0f + __expf(-10.0f * z));
}

// generic -> LDS(addrspace 3) 32-bit byte address, for async-to-LDS vdst operand.
// Two-step cast (ptr -> u64 -> u32) so the host pass also type-checks.
__device__ __forceinline__ unsigned lds_addr32(const void* p) {
    return (unsigned)(unsigned long long)(__attribute__((address_space(3))) const void*)p;
}

__global__ __launch_bounds__(32)
void racnn_crop_resize_wmma(const float* __restrict__ img,
                            const float* __restrict__ locs,
                            float* __restrict__ out)
{
    // LDS: padded to 33 so strided (stride-33) per-M reads hit distinct banks
    __shared__ float A_s[16][33];   // row-interp weights * row mask   (16 x 32 used)
    __shared__ float P_s[32][33];   // 32x32 input patch               (32 x 32 used)
    __shared__ float W_s[32][16];   // col-interp weights * col mask   (32 x 16)
    __shared__ float T_s[32][16];   // temp = P @ W

    const int lane = threadIdx.x;          // 0..31 (wave32)
    const int half = lane >> 4;            // 0: lanes 0-15, 1: lanes 16-31
    const int l16  = lane & 15;

    const int b  = blockIdx.z;
    const int c  = blockIdx.y;
    const int i0 = (blockIdx.x / 14) * 16; // output row tile base
    const int j0 = (blockIdx.x % 14) * 16; // output col tile base

    // ---- per-sample crop box (uniform across the wave) ----
    float tx = locs[b * 3 + 0];
    float ty = locs[b * 3 + 1];
    float tl = locs[b * 3 + 2];
    tl = fmaxf(tl, (float)SS / 3.0f);
    tx = fminf(fmaxf(tx, tl), (float)SS - tl);
    ty = fminf(fmaxf(ty, tl), (float)SS - tl);
    const float w_off = floorf(fmaxf(tx - tl, 0.0f));
    const float h_off = floorf(fmaxf(ty - tl, 0.0f));
    const float w_end = fminf(floorf(tx + tl), (float)SS);
    const float h_end = fminf(floorf(ty + tl), (float)SS);
    const float sr = (w_end - w_off - 1.0f) / (float)(OUTN - 1); // row scale (<=1)
    const float sc = (h_end - h_off - 1.0f) / (float)(OUTN - 1); // col scale (<=1)

    // ---- base input row/col of the 32-wide window covering this tile ----
    float rr0 = w_off + (float)i0 * sr;
    int rbase = (int)fminf(fmaxf(floorf(rr0), 0.0f), (float)(SS - 1));
    if (rbase > SS - 32) rbase = SS - 32;
    float cc0 = h_off + (float)j0 * sc;
    int cbase = (int)fminf(fmaxf(floorf(cc0), 0.0f), (float)(SS - 1));
    if (cbase > SS - 32) cbase = SS - 32;

    // ---- kick off async DMA of the 32x32 patch straight into LDS ----
    // One GLOBAL_LOAD_ASYNC_TO_LDS_B32 per patch row: 32 lanes x 4B = 128B
    // coalesced, no VGPR landing zone, tracked by ASYNCcnt. Overlaps with the
    // transcendental-heavy weight construction below.
    {
        const float* gp   = img + (((size_t)b * 3 + c) * SS + rbase) * SS + cbase + lane;
        const unsigned lp = lds_addr32(&P_s[0][lane]);
#pragma unroll
        for (int r = 0; r < 32; ++r) {
            asm volatile("global_load_async_to_lds_b32 %0, %1, off"
                         :
                         : "v"(lp + (unsigned)(r * 33 * 4)), "v"(gp + (size_t)r * SS)
                         : "memory");
        }
    }

    // ---- zero weight matrices ----
    for (int t = lane; t < 16 * 33; t += 32) (&A_s[0][0])[t] = 0.0f;
    for (int t = lane; t < 32 * 16; t += 32) (&W_s[0][0])[t] = 0.0f;
    __syncthreads();

    // ---- build interpolation+mask weights (2 nonzeros per output row/col) ----
    if (lane < 16) {
        const int i = i0 + lane;
        const float rr = w_off + (float)i * sr;
        int r0 = (int)fminf(fmaxf(floorf(rr), 0.0f), (float)(SS - 1));
        int r1 = min(r0 + 1, SS - 1);
        const float fr = rr - (float)r0;
        const float m0 = sig10((float)r0 - w_off) - sig10((float)r0 - w_end);
        const float m1 = sig10((float)r1 - w_off) - sig10((float)r1 - w_end);
        A_s[lane][r0 - rbase] += (1.0f - fr) * m0;
        A_s[lane][r1 - rbase] += fr * m1;
    } else {
        const int j = j0 + l16;
        const float cc = h_off + (float)j * sc;
        int c0 = (int)fminf(fmaxf(floorf(cc), 0.0f), (float)(SS - 1));
        int c1 = min(c0 + 1, SS - 1);
        const float fc = cc - (float)c0;
        const float m0 = sig10((float)c0 - h_off) - sig10((float)c0 - h_end);
        const float m1 = sig10((float)c1 - h_off) - sig10((float)c1 - h_end);
        W_s[c0 - cbase][l16] += (1.0f - fc) * m0;
        W_s[c1 - cbase][l16] += fc * m1;
    }

    // ---- drain the async patch DMA, then sync the workgroup ----
    asm volatile("s_wait_asynccnt 0" ::: "memory");
    __syncthreads();

    // ---- GEMM 1: T(32x16) = P(32x32) @ W(32x16), K-steps of 4 ----
    v8f accT = {0.f,0.f,0.f,0.f,0.f,0.f,0.f,0.f};
    v8f accB = {0.f,0.f,0.f,0.f,0.f,0.f,0.f,0.f};
#pragma unroll
    for (int kk = 0; kk < 8; ++kk) {
        const int k0 = kk * 4;
        v2f aT, aB, bb;
        // A operand (16x4): .x -> K=k0 (lanes0-15) / k0+2 (lanes16-31); .y -> k0+1 / k0+3
        aT.x = P_s[l16][k0 + 2 * half];
        aT.y = P_s[l16][k0 + 2 * half + 1];
        aB.x = P_s[16 + l16][k0 + 2 * half];
        aB.y = P_s[16 + l16][k0 + 2 * half + 1];
        // B operand (4x16): .x -> row k0 / k0+2; .y -> row k0+1 / k0+3
        bb.x = W_s[k0 + 2 * half][l16];
        bb.y = W_s[k0 + 2 * half + 1][l16];
        accT = __builtin_amdgcn_wmma_f32_16x16x4_f32(false, aT, false, bb,
                                                     (short)0, accT, false, false);
        accB = __builtin_amdgcn_wmma_f32_16x16x4_f32(false, aB, false, bb,
                                                     (short)0, accB, false, false);
    }

    // spill T to LDS in row-major (C/D layout: acc[v] holds row v + 8*half)
#pragma unroll
    for (int v = 0; v < 8; ++v) {
        T_s[v + 8 * half][l16]      = accT[v];
        T_s[16 + v + 8 * half][l16] = accB[v];
    }
    __syncthreads();

    // ---- GEMM 2: out(16x16) = A(16x32) @ T(32x16) ----
    v8f acc = {0.f,0.f,0.f,0.f,0.f,0.f,0.f,0.f};
#pragma unroll
    for (int kk = 0; kk < 8; ++kk) {
        const int k0 = kk * 4;
        v2f aa, bb;
        aa.x = A_s[l16][k0 + 2 * half];
        aa.y = A_s[l16][k0 + 2 * half + 1];
        bb.x = T_s[k0 + 2 * half][l16];
        bb.y = T_s[k0 + 2 * half + 1][l16];
        acc = __builtin_amdgcn_wmma_f32_16x16x4_f32(false, aa, false, bb,
                                                    (short)0, acc, false, false);
    }

    // ---- store output tile (rows i0+v+8*half, cols j0+l16) ----
    float* op = out + (((size_t)b * 3 + c) * OUTN + i0) * OUTN + j0;
#pragma unroll
    for (int v = 0; v < 8; ++v) {
        op[(size_t)(v + 8 * half) * OUTN + l16] = acc[v];
    }
}

extern "C" void kernel_launch(void* const* d_in, const int* in_sizes, int n_in,
                              void* d_out, int out_size, void* d_ws, size_t ws_size,
                              hipStream_t stream) {
    const float* images = (const float*)d_in[0]; // [B,3,224,224] fp32
    const float* locs   = (const float*)d_in[1]; // [B,3] fp32
    float* out          = (float*)d_out;         // [B,3,224,224] fp32

    const int B = in_sizes[1] / 3;
    dim3 grid(14 * 14, 3, B);   // 196 tiles x 3 channels x B samples
    dim3 block(32);             // one wave32 per 16x16 output tile
    racnn_crop_resize_wmma<<<grid, block, 0, stream>>>(images, locs, out);
}